// Quanvolution_65481071398013
// MI455X (gfx1250) — compile-verified
//
#include <hip/hip_runtime.h>
#include <math.h>

typedef __attribute__((ext_vector_type(2))) float v2f;
typedef __attribute__((ext_vector_type(4))) float v4f;
typedef __attribute__((ext_vector_type(8))) float v8f;

// ============================================================================
// Kernel 1: patches -> embedding angles -> 4-qubit circuit sim -> Z features
// One thread per shot; full 16-amplitude complex state lives in registers.
// Wire w pairs amplitudes at stride (8 >> w)  (wire 0 = MSB of the 4-bit index).
// ============================================================================

#define RY_GATE(MASK, C, S)                                               \
  _Pragma("unroll")                                                       \
  for (int i = 0; i < 16; ++i)                                            \
    if (!(i & (MASK))) {                                                  \
      int j = i | (MASK);                                                 \
      float ar = re[i], ai = im[i], br = re[j], bi = im[j];               \
      re[i] = (C)*ar - (S)*br;  im[i] = (C)*ai - (S)*bi;                  \
      re[j] = (S)*ar + (C)*br;  im[j] = (S)*ai + (C)*bi;                  \
    }

#define RX_GATE(MASK, C, S)                                               \
  _Pragma("unroll")                                                       \
  for (int i = 0; i < 16; ++i)                                            \
    if (!(i & (MASK))) {                                                  \
      int j = i | (MASK);                                                 \
      float ar = re[i], ai = im[i], br = re[j], bi = im[j];               \
      re[i] = (C)*ar + (S)*bi;  im[i] = (C)*ai - (S)*br;                  \
      re[j] = (S)*ai + (C)*br;  im[j] = (C)*bi - (S)*ar;                  \
    }

#define RZ_GATE(MASK, C, S)                                               \
  _Pragma("unroll")                                                       \
  for (int i = 0; i < 16; ++i) {                                          \
    float sg = (i & (MASK)) ? (S) : -(S);                                 \
    float r0 = re[i], q0 = im[i];                                         \
    re[i] = (C)*r0 - sg*q0;  im[i] = (C)*q0 + sg*r0;                      \
  }

#define CNOT_GATE(CM, TM)                                                 \
  _Pragma("unroll")                                                       \
  for (int i = 0; i < 16; ++i)                                            \
    if ((i & (CM)) && !(i & (TM))) {                                      \
      int j = i | (TM);                                                   \
      float t;                                                            \
      t = re[i]; re[i] = re[j]; re[j] = t;                                \
      t = im[i]; im[i] = im[j]; im[j] = t;                                \
    }

__global__ __launch_bounds__(256) void quanv_kernel(
    const float* __restrict__ x, const float* __restrict__ embW,
    const float* __restrict__ embB, const float* __restrict__ qp,
    float* __restrict__ feats, int nShots)
{
  int tid = blockIdx.x * blockDim.x + threadIdx.x;
  if (tid >= nShots) return;

  int b  = tid / 196;
  int p  = tid - b * 196;
  int py = p / 14;
  int px = p - py * 14;
  const float* img = x + (size_t)b * 784 + (size_t)(2 * py) * 28 + 2 * px;
  float patch[4] = { img[0], img[1], img[28], img[29] };

  // Embedding angles + RY-layer half-angle sincos.
  float cs[4], sn[4];
#pragma unroll
  for (int w = 0; w < 4; ++w) {
    float ang = embB[w];
#pragma unroll
    for (int k = 0; k < 4; ++k) ang = fmaf(patch[k], embW[w * 4 + k], ang);
    __sincosf(0.5f * ang, &sn[w], &cs[w]);
  }

  // |0000> after 4 RY gates is a real product state.
  float re[16], im[16];
#pragma unroll
  for (int i = 0; i < 16; ++i) {
    float a0 = (i & 8) ? sn[0] : cs[0];
    float a1 = (i & 4) ? sn[1] : cs[1];
    float a2 = (i & 2) ? sn[2] : cs[2];
    float a3 = (i & 1) ? sn[3] : cs[3];
    re[i] = a0 * a1 * a2 * a3;
    im[i] = 0.0f;
  }

  // Parameterized layer (q_params are wave-uniform; sincos cost is trivial).
  float qc[5], qs[5];
#pragma unroll
  for (int j = 0; j < 5; ++j) __sincosf(0.5f * qp[j], &qs[j], &qc[j]);

  RX_GATE(8, qc[0], qs[0])      // RX(q0) wire 0
  RY_GATE(4, qc[1], qs[1])      // RY(q1) wire 1
  RZ_GATE(2, qc[2], qs[2])      // RZ(q2) wire 2
  CNOT_GATE(8, 4)               // CNOT(0,1)
  RX_GATE(1, qc[3], qs[3])      // RX(q3) wire 3
  CNOT_GATE(2, 1)               // CNOT(2,3)
  RY_GATE(2, qc[4], qs[4])      // RY(q4) wire 2
  CNOT_GATE(4, 2)               // CNOT(1,2)

  // Z expectation per wire.
  float ez[4];
#pragma unroll
  for (int w = 0; w < 4; ++w) {
    int m = 8 >> w;
    float acc = 0.0f;
#pragma unroll
    for (int i = 0; i < 16; ++i) {
      float pr = re[i] * re[i] + im[i] * im[i];
      acc += (i & m) ? -pr : pr;
    }
    ez[w] = acc;
  }
  v4f outv = { ez[0], ez[1], ez[2], ez[3] };
  *(v4f*)(feats + (size_t)tid * 4) = outv;   // feats[b*784 + p*4 + w]
}

// ============================================================================
// Kernel 2: logits = feats(2048x784) @ lin_W^T(784x10) + b, then log_softmax,
// via V_WMMA_F32_16X16X4_F32 (full f32 path). One wave per 16 rows; N padded
// to 16; 196 WMMA issues per wave over K=784. lin_W^T staged in LDS.
// A frag: lane<16 -> (M=lane, K 0..1), lane>=16 -> (M=lane-16, K 2..3); B mirrored.
// ============================================================================
#define BSTRIDE 786   // even -> 8B-aligned ds_load_b64; 786 mod 64 = 18 (2-way)

__global__ __launch_bounds__(32) void head_kernel(
    const float* __restrict__ feats, const float* __restrict__ linW,
    const float* __restrict__ linB, float* __restrict__ out, int B)
{
  __shared__ float Bl[16 * BSTRIDE];

  const int lane = threadIdx.x;
  // Stage B = lin_W^T padded (cols 10..15 = 0) into LDS.
  for (int idx = lane; idx < 16 * 784; idx += 32) {
    int n = idx / 784;
    int k = idx - n * 784;
    Bl[n * BSTRIDE + k] = (n < 10) ? linW[n * 784 + k] : 0.0f;
  }
  __syncthreads();

  const int row0 = blockIdx.x * 16;
  const int l16  = lane & 15;
  const int hi   = lane >> 4;       // 0: K pair {0,1}; 1: K pair {2,3}
  int row = row0 + l16;
  if (row >= B) row = B - 1;        // inert for B % 16 == 0; keeps EXEC uniform
  const float* arow = feats + (size_t)row * 784;
  const float* brow = &Bl[l16 * BSTRIDE];
  const int ko = hi * 2;

  v8f acc = {};
  for (int k = 0; k < 784; k += 4) {
    v2f a  = *(const v2f*)(arow + k + ko);
    v2f bf = *(const v2f*)(brow + k + ko);
    // (neg_a, A, neg_b, B, c_mod, C, reuse_a, reuse_b)
    acc = __builtin_amdgcn_wmma_f32_16x16x4_f32(
        false, a, false, bf, (short)0, acc, false, false);
  }

  // D layout: VGPR r, lanes 0-15 -> row (row0+r), lanes 16-31 -> row (row0+8+r),
  // N = l16. Each row's 16 logits live in one 16-lane half.
  const bool  valid = (l16 < 10);
  const float bias  = valid ? linB[l16] : 0.0f;

#pragma unroll
  for (int r = 0; r < 8; ++r) {
    float v  = acc[r] + bias;
    float vm = valid ? v : -3.4e38f;
#pragma unroll
    for (int off = 8; off > 0; off >>= 1)
      vm = fmaxf(vm, __shfl_xor(vm, off, 16));
    float e = valid ? __expf(v - vm) : 0.0f;
    float ssum = e;
#pragma unroll
    for (int off = 8; off > 0; off >>= 1)
      ssum += __shfl_xor(ssum, off, 16);
    float resv = v - vm - __logf(ssum);
    int orow = row0 + r + hi * 8;
    if (valid && orow < B) out[orow * 10 + l16] = resv;
  }
}

// ============================================================================
extern "C" void kernel_launch(void* const* d_in, const int* in_sizes, int n_in,
                              void* d_out, int out_size, void* d_ws, size_t ws_size,
                              hipStream_t stream)
{
  const float* x    = (const float*)d_in[0];
  const float* embW = (const float*)d_in[1];
  const float* embB = (const float*)d_in[2];
  const float* qp   = (const float*)d_in[3];
  const float* linW = (const float*)d_in[4];
  const float* linB = (const float*)d_in[5];
  float* out = (float*)d_out;

  const int B      = in_sizes[0] / 784;   // 2048
  const int nShots = B * 196;             // 401408
  float* feats = (float*)d_ws;            // B*784 floats = 6.4 MB scratch

  quanv_kernel<<<dim3((nShots + 255) / 256), dim3(256), 0, stream>>>(
      x, embW, embB, qp, feats, nShots);

  head_kernel<<<dim3((B + 15) / 16), dim3(32), 0, stream>>>(
      feats, linW, linB, out, B);
}